// DihedralAngleLayer_54769422959211
// MI455X (gfx1250) — compile-verified
//
#include <hip/hip_runtime.h>
#include <math.h>

// Butane dihedral: B rows x 42 f32 in, B f32 out. Pure streaming kernel.
// Strategy: async-copy 256 contiguous rows (43008 B) per block into LDS via
// CDNA5 global_load_async_to_lds_b128 (ASYNCcnt), then one lane per row.

#define TPB        256   // 8 wave32 waves
#define ROWS       256   // rows per block == one row per thread
#define ROW_FLOATS 42
#define ROW_BYTES  168   // 42 * 4

__global__ __launch_bounds__(TPB)
void dihedral_kernel(const float* __restrict__ x, float* __restrict__ out, int B) {
    __shared__ float smem[ROWS * ROW_FLOATS];   // 43008 bytes

    const int tid = threadIdx.x;
    const long long base_row = (long long)blockIdx.x * ROWS;
    int rows = B - (int)base_row;
    if (rows > ROWS) rows = ROWS;
    const int T = rows * ROW_BYTES;             // bytes to stage for this block

    const char* gsrc = (const char*)x + base_row * (long long)ROW_BYTES;
    const unsigned lds_base = (unsigned)(size_t)(void*)smem;  // low 32 bits = LDS byte offset

    // ---- async copy: global -> LDS, 16 B per lane per issue (4096 B/iter) ----
    const int nfull = T & ~15;
    for (int off = tid * 16; off < nfull; off += TPB * 16) {
        unsigned lds = lds_base + (unsigned)off;
        const char* g = gsrc + off;
        asm volatile("global_load_async_to_lds_b128 %0, %1, off"
                     :: "v"(lds), "v"(g) : "memory");
    }
    // 4-byte tail (T always a multiple of 8; of 16 for even row counts)
    for (int off = nfull + tid * 4; off < T; off += TPB * 4) {
        unsigned lds = lds_base + (unsigned)off;
        const char* g = gsrc + off;
        asm volatile("global_load_async_to_lds_b32 %0, %1, off"
                     :: "v"(lds), "v"(g) : "memory");
    }

#if __has_builtin(__builtin_amdgcn_s_wait_asynccnt)
    __builtin_amdgcn_s_wait_asynccnt(0);
#else
    asm volatile("s_wait_asynccnt 0x0" ::: "memory");
#endif
    __syncthreads();

    // ---- per-row dihedral from LDS ----
    if (tid < rows) {
        const float* r = &smem[tid * ROW_FLOATS];
        // carbons at atom indices 0,4,7,11 -> float offsets 0,12,21,33
        float c0x = r[0],  c0y = r[1],  c0z = r[2];
        float c1x = r[12], c1y = r[13], c1z = r[14];
        float c2x = r[21], c2y = r[22], c2z = r[23];
        float c3x = r[33], c3y = r[34], c3z = r[35];

        float v0x = c1x - c0x, v0y = c1y - c0y, v0z = c1z - c0z;
        float v1x = c2x - c1x, v1y = c2y - c1y, v1z = c2z - c1z;
        float v2x = c3x - c2x, v2y = c3y - c2y, v2z = c3z - c2z;

        // na = (-v0) x v1
        float nax = v0z * v1y - v0y * v1z;
        float nay = v0x * v1z - v0z * v1x;
        float naz = v0y * v1x - v0x * v1y;
        // nb = (-v1) x v2
        float nbx = v1z * v2y - v1y * v2z;
        float nby = v1x * v2z - v1z * v2x;
        float nbz = v1y * v2x - v1x * v2y;

        float xx = nax * nbx + nay * nby + naz * nbz;

        // xp = na x nb
        float xpx = nay * nbz - naz * nby;
        float xpy = naz * nbx - nax * nbz;
        float xpz = nax * nby - nay * nbx;

        float n1 = sqrtf(v1x * v1x + v1y * v1y + v1z * v1z);
        float yy = (v1x * xpx + v1y * xpy + v1z * xpz) / n1;

        out[base_row + tid] = atan2f(yy, xx);
    }
}

extern "C" void kernel_launch(void* const* d_in, const int* in_sizes, int n_in,
                              void* d_out, int out_size, void* d_ws, size_t ws_size,
                              hipStream_t stream) {
    const float* x = (const float*)d_in[0];     // [B, 42] f32
    // d_in[1] is the fixed one-hot mask (carbons 0,4,7,11) -> folded into offsets
    float* out = (float*)d_out;                  // [B] f32

    const int B = in_sizes[0] / (3 * 14);
    const int grid = (B + ROWS - 1) / ROWS;
    dihedral_kernel<<<grid, TPB, 0, stream>>>(x, out, B);
}